// NmsDecoder_31937376813470
// MI455X (gfx1250) — compile-verified
//
#include <hip/hip_runtime.h>
#include <math.h>

#define B_      8
#define N_      16384
#define C_      80
#define CAP_    2048
#define KCAND_  256
#define MAXPC_  100
#define MAXDET_ 100
#define SCORE_THR_ 0.5f

typedef float v2f __attribute__((ext_vector_type(2)));
typedef float v8f __attribute__((ext_vector_type(8)));

__device__ __forceinline__ bool before_desc(float ka, unsigned ma, float kb, unsigned mb) {
  return (ka > kb) || ((ka == kb) && (ma < mb));
}

// Bitonic sort of 2048 (key desc, meta asc) with 256 threads.
__device__ void bitonic2048(float* key, unsigned* val, int tid) {
  for (unsigned k = 2; k <= 2048u; k <<= 1) {
    for (unsigned j = k >> 1; j > 0; j >>= 1) {
      __syncthreads();
      for (unsigned t = (unsigned)tid; t < 1024u; t += 256u) {
        unsigned i = 2u * j * (t / j) + (t % j);
        unsigned p = i + j;
        float ki = key[i], kp = key[p];
        unsigned mi = val[i], mp = val[p];
        bool up = ((i & k) == 0);
        bool sw = up ? before_desc(kp, mp, ki, mi)
                     : before_desc(ki, mi, kp, mp);
        if (sw) { key[i] = kp; key[p] = ki; val[i] = mp; val[p] = mi; }
      }
    }
  }
  __syncthreads();
}

__global__ void k_init(float* out, int outN, int* cnt, int* kcnt) {
  int g = blockIdx.x * blockDim.x + threadIdx.x;
  if (g < outN) out[g] = 0.0f;
  if (g < B_ * C_) { cnt[g] = 0; kcnt[g] = 0; }
}

// One pass over logits: softmax normalizer + argmax; only argmax can have p > 0.5.
__global__ void k_softmax_filter(const float* __restrict__ conf,
                                 int* __restrict__ cnt,
                                 float* __restrict__ candScore,
                                 unsigned* __restrict__ candIdx) {
  int g = blockIdx.x * blockDim.x + threadIdx.x;
  if (g >= B_ * N_) return;
  const float4* row = (const float4*)(conf + (size_t)g * C_);
  float m = -INFINITY; int am = 0;
#pragma unroll
  for (int q = 0; q < C_ / 4; ++q) {
    float4 v = row[q];
    if (v.x > m) { m = v.x; am = q * 4 + 0; }
    if (v.y > m) { m = v.y; am = q * 4 + 1; }
    if (v.z > m) { m = v.z; am = q * 4 + 2; }
    if (v.w > m) { m = v.w; am = q * 4 + 3; }
  }
  float s = 0.f;
#pragma unroll
  for (int q = 0; q < C_ / 4; ++q) {
    float4 v = row[q];
    s += expf(v.x - m) + expf(v.y - m) + expf(v.z - m) + expf(v.w - m);
  }
  float p = 1.0f / s;  // softmax prob of the argmax class
  if (p > SCORE_THR_) {
    int b = g / N_;
    int n = g - b * N_;
    int slot = b * C_ + am;
    int pos = atomicAdd(&cnt[slot], 1);
    if (pos < CAP_) {
      candScore[(size_t)slot * CAP_ + pos] = p;
      candIdx[(size_t)slot * CAP_ + pos]   = (unsigned)n;
    }
  }
}

// Per (image,class) slot: sort candidates, top-256, IoU mask (WMMA area-sum +
// VALU intersection), greedy NMS, emit up to 100 kept (sorted by score desc).
__global__ void __launch_bounds__(256) k_nms_class(
    const float* __restrict__ box_pred,
    const int* __restrict__ cnt,
    const float* __restrict__ candScore,
    const unsigned* __restrict__ candIdx,
    float* __restrict__ keepScore,
    unsigned* __restrict__ keepMeta,
    int* __restrict__ kcount) {
  __shared__ float    sKey[CAP_];
  __shared__ unsigned sVal[CAP_];
  __shared__ float sY1[KCAND_], sX1[KCAND_], sY2[KCAND_], sX2[KCAND_], sA[KCAND_];
  __shared__ unsigned sMask[KCAND_ * 8];

  const int tid  = threadIdx.x;
  const int slot = blockIdx.x;
  const int b = slot / C_;
  const int c = slot % C_;

  int n = cnt[slot]; if (n > CAP_) n = CAP_;

  for (int i = tid; i < CAP_; i += 256) {
    if (i < n) {
      sKey[i] = candScore[(size_t)slot * CAP_ + i];
      sVal[i] = candIdx[(size_t)slot * CAP_ + i];
    } else { sKey[i] = -1.0f; sVal[i] = 0xFFFFFFFFu; }
  }
  bitonic2048(sKey, sVal, tid);

  const int nv = (n < KCAND_) ? n : KCAND_;
  if (tid < KCAND_) {
    float y1 = 0, x1 = 0, y2 = 0, x2 = 0, ar = 0;
    if (tid < nv) {
      unsigned idx = sVal[tid];
      const float* bp = box_pred + ((size_t)b * N_ + idx) * 4;
      float x = bp[0], y = bp[1], w = bp[2], h = bp[3];
      y1 = y; x1 = x; y2 = y + h; x2 = x + w;
      ar = (y2 - y1) * (x2 - x1);
    }
    sY1[tid] = y1; sX1[tid] = x1; sY2[tid] = y2; sX2[tid] = x2; sA[tid] = ar;
  }
  for (int i = tid; i < KCAND_ * 8; i += 256) sMask[i] = 0u;
  __syncthreads();

  // 16x16 tiles of the 256x256 pair space. WMMA computes D[i][j] = a_i + a_j
  // (A rows = [a_i, 1, 0, 0]; B cols = [1, a_j, 0, 0]); VALU computes inter.
  // iou > 0.5  <=>  3*inter > a_i + a_j.
  const int lane = tid & 31;
  const int wave = tid >> 5;
  const int half = lane >> 4;
  const int l15  = lane & 15;
  for (int tile = wave; tile < 256; tile += 8) {
    const int ti = tile >> 4, tj = tile & 15;
    v2f a, bb;
    if (half == 0) {
      a[0]  = sA[ti * 16 + l15]; a[1]  = 1.0f;   // K=0,1 for row M=l15
      bb[0] = 1.0f;              bb[1] = sA[tj * 16 + l15];
    } else {
      a[0] = 0.f; a[1] = 0.f; bb[0] = 0.f; bb[1] = 0.f;  // K=2,3 unused
    }
    v8f cz = {};
    v8f dsum = __builtin_amdgcn_wmma_f32_16x16x4_f32(
        false, a, false, bb, (short)0, cz, false, false);
    const int j = tj * 16 + l15;
    const float jy1 = sY1[j], jx1 = sX1[j], jy2 = sY2[j], jx2 = sX2[j];
#pragma unroll
    for (int r = 0; r < 8; ++r) {
      const int i = ti * 16 + r + 8 * half;  // C/D layout: VGPR r, lane half
      float ih = fminf(sY2[i], jy2) - fmaxf(sY1[i], jy1);
      float iw = fminf(sX2[i], jx2) - fmaxf(sX1[i], jx1);
      float inter = fmaxf(ih, 0.f) * fmaxf(iw, 0.f);
      if (3.0f * inter > dsum[r])
        atomicOr(&sMask[i * 8 + (j >> 5)], 1u << (j & 31));
    }
  }
  __syncthreads();

  if (tid == 0) {
    unsigned keep[8];
#pragma unroll
    for (int w = 0; w < 8; ++w) {
      int lo = w * 32, hi = lo + 32;
      keep[w] = (nv >= hi) ? 0xFFFFFFFFu : ((nv <= lo) ? 0u : ((1u << (nv - lo)) - 1u));
    }
    for (int i = 0; i < KCAND_; ++i) {
      if ((keep[i >> 5] >> (i & 31)) & 1u) {
#pragma unroll
        for (int w = 0; w < 8; ++w) {
          unsigned m = sMask[i * 8 + w];
          if (w < (i >> 5)) m = 0u;                                      // only j > i
          else if (w == (i >> 5)) m &= ~(unsigned)((1ull << ((i & 31) + 1)) - 1ull);
          keep[w] &= ~m;
        }
      }
    }
    int kc = 0;
    for (int i = 0; i < KCAND_ && kc < MAXPC_; ++i) {
      if ((keep[i >> 5] >> (i & 31)) & 1u) {
        keepScore[(size_t)slot * MAXPC_ + kc] = sKey[i];
        // meta ascending == reference flat index c*256+i ascending (tie-break)
        keepMeta[(size_t)slot * MAXPC_ + kc] =
            ((unsigned)c << 22) | ((unsigned)i << 14) | (sVal[i] & 0x3FFFu);
        kc++;
      }
    }
    kcount[slot] = kc;
  }
}

// Per image: 100-step merge of 80 sorted per-class lists via block argmax.
__global__ void __launch_bounds__(256) k_final_topk(
    const float* __restrict__ box_pred,
    const float* __restrict__ keepScore,
    const unsigned* __restrict__ keepMeta,
    const int* __restrict__ kcount,
    float* __restrict__ out) {
  __shared__ int ptr[C_];
  __shared__ float    redK[256];
  __shared__ unsigned redM[256];
  __shared__ int      redC[256];
  __shared__ int numKept;

  const int tid = threadIdx.x;
  const int b   = blockIdx.x;
  if (tid < C_) ptr[tid] = 0;
  if (tid == 0) numKept = 0;
  __syncthreads();

  float* outB = out;                          // B*100*4 (xywh)
  float* outS = out + B_ * MAXDET_ * 4;       // B*100
  float* outC = outS + B_ * MAXDET_;          // B*100
  float* outN = outC + B_ * MAXDET_;          // B

  for (int it = 0; it < MAXDET_; ++it) {
    float k = -1.0f; unsigned m = 0xFFFFFFFFu; int cc = -1;
    if (tid < C_) {
      int slot = b * C_ + tid;
      int p = ptr[tid];
      if (p < kcount[slot]) {
        k = keepScore[(size_t)slot * MAXPC_ + p];
        m = keepMeta[(size_t)slot * MAXPC_ + p];
        cc = tid;
      }
    }
    redK[tid] = k; redM[tid] = m; redC[tid] = cc;
    __syncthreads();
    for (int s = 128; s >= 1; s >>= 1) {
      if (tid < s) {
        if (before_desc(redK[tid + s], redM[tid + s], redK[tid], redM[tid])) {
          redK[tid] = redK[tid + s]; redM[tid] = redM[tid + s]; redC[tid] = redC[tid + s];
        }
      }
      __syncthreads();
    }
    if (tid == 0) {
      float wk = redK[0]; unsigned wm = redM[0]; int wc = redC[0];
      if (wk > 0.0f && wc >= 0) {
        unsigned idx = wm & 0x3FFFu;
        const float* bp = box_pred + ((size_t)b * N_ + idx) * 4;
        float x = bp[0], y = bp[1], w = bp[2], h = bp[3];
        float x2 = x + w, y2 = y + h;             // reproduce reference rounding
        float* ob = outB + ((size_t)b * MAXDET_ + it) * 4;
        ob[0] = x; ob[1] = y; ob[2] = x2 - x; ob[3] = y2 - y;
        outS[b * MAXDET_ + it] = wk;
        outC[b * MAXDET_ + it] = (float)(wm >> 22);
        ptr[wc] = ptr[wc] + 1;
        numKept++;
      } // else: slots already zeroed by k_init
    }
    __syncthreads();
  }
  if (tid == 0) outN[b] = (float)numKept;
}

extern "C" void kernel_launch(void* const* d_in, const int* in_sizes, int n_in,
                              void* d_out, int out_size, void* d_ws, size_t ws_size,
                              hipStream_t stream) {
  (void)in_sizes; (void)n_in; (void)ws_size;
  const float* box  = (const float*)d_in[0];  // (B,N,4) xywh
  const float* conf = (const float*)d_in[1];  // (B,N,C) logits
  float* out = (float*)d_out;                 // 3200 + 800 + 800 + 8 floats

  char* ws = (char*)d_ws;
  int* cnt  = (int*)ws;                  ws += (size_t)B_ * C_ * sizeof(int);
  int* kcnt = (int*)ws;                  ws += (size_t)B_ * C_ * sizeof(int);
  float*    candScore = (float*)ws;      ws += (size_t)B_ * C_ * CAP_ * sizeof(float);
  unsigned* candIdx   = (unsigned*)ws;   ws += (size_t)B_ * C_ * CAP_ * sizeof(unsigned);
  float*    keepScoreA = (float*)ws;     ws += (size_t)B_ * C_ * MAXPC_ * sizeof(float);
  unsigned* keepMetaA  = (unsigned*)ws;  ws += (size_t)B_ * C_ * MAXPC_ * sizeof(unsigned);

  k_init<<<(out_size + 255) / 256, 256, 0, stream>>>(out, out_size, cnt, kcnt);
  k_softmax_filter<<<(B_ * N_) / 256, 256, 0, stream>>>(conf, cnt, candScore, candIdx);
  k_nms_class<<<B_ * C_, 256, 0, stream>>>(box, cnt, candScore, candIdx,
                                           keepScoreA, keepMetaA, kcnt);
  k_final_topk<<<B_, 256, 0, stream>>>(box, keepScoreA, keepMetaA, kcnt, out);
}